// Attention_4260607558096
// MI455X (gfx1250) — compile-verified
//
#include <hip/hip_runtime.h>
#include <hip/hip_bf16.h>

// ---------------------------------------------------------------------------
// Attention prefill (B=8, SEQ=16, D=4096, H=32, KVH=8, HD=128, START=1024)
// fp32 end-to-end, all matmuls via V_WMMA_F32_16X16X4_F32 (wave32, 16x16 tiles)
// ---------------------------------------------------------------------------

#define Dm       4096
#define Hh       32
#define KVH      8
#define HD       128
#define Bb       8
#define SEQ      16
#define MAXLEN   2048
#define STARTP   1024
#define KLEN     (STARTP + SEQ)     // 1040 = 65 * 16
#define MROWS    (Bb * SEQ)         // 128

typedef __attribute__((ext_vector_type(2))) float v2f;
typedef __attribute__((ext_vector_type(8))) float v8f;

// D = A(16x4) * B(4x16) + C  -- fp32 WMMA, exact-precision accumulate
__device__ __forceinline__ v8f wmma_f32(v2f a, v2f b, v8f c) {
    return __builtin_amdgcn_wmma_f32_16x16x4_f32(
        /*neg_a=*/false, a, /*neg_b=*/false, b,
        /*c_mod=*/(short)0, c, /*reuse_a=*/false, /*reuse_b=*/false);
}

// ---------------------------------------------------------------------------
// Generic GEMM: C(128 x N) = A(128 x K) @ W(K x N), all row-major fp32.
// Block = 128 threads (4 waves). Block tile = 16 (M) x 256 (N).
// Each wave owns 4 adjacent 16x16 output tiles (4 independent WMMA chains).
// A K-slab (16 x 32) is staged in padded LDS (stride 36 floats -> 16B aligned,
// bank-conflict-free A-fragment reads).
// grid = (N/256, 128/16)
// ---------------------------------------------------------------------------
#define GEMM_BK 32

__global__ void gemm128_wmma(const float* __restrict__ A,
                             const float* __restrict__ W,
                             float* __restrict__ C,
                             int K, int N) {
    __shared__ alignas(16) float As[16][GEMM_BK + 4];   // stride 36 floats

    const int tid  = threadIdx.x;
    const int wave = tid >> 5;
    const int lane = tid & 31;
    const int m0   = blockIdx.y * 16;
    const int nw   = blockIdx.x * 256 + wave * 64;      // wave's first column

    const int mrow = lane & 15;
    const int koff = (lane >> 4) * 2;                   // lanes 0-15: K0/K1, 16-31: K2/K3
    const int nc   = nw + mrow;                         // this lane's N column

    // A staging coords: thread -> one float4 of the 16x32 slab
    const int sr = tid >> 3;            // row 0..15
    const int sc = (tid & 7) * 4;       // col 0,4,...,28

    v8f acc0 = {}, acc1 = {}, acc2 = {}, acc3 = {};

    for (int k0 = 0; k0 < K; k0 += GEMM_BK) {
        // cooperative coalesced stage of A[m0:m0+16, k0:k0+32]
        const float4 a4 = *reinterpret_cast<const float4*>(
            &A[(size_t)(m0 + sr) * K + k0 + sc]);
        *reinterpret_cast<float4*>(&As[sr][sc]) = a4;
        if (k0 + GEMM_BK < K) {
            __builtin_prefetch(&A[(size_t)(m0 + sr) * K + k0 + GEMM_BK + sc], 0, 1);
        }
        __syncthreads();

        #pragma unroll
        for (int kk = 0; kk < GEMM_BK; kk += 4) {
            // A fragment from LDS (rows striped across lanes, K across VGPRs)
            v2f a;
            a.x = As[mrow][kk + koff];
            a.y = As[mrow][kk + koff + 1];
            // B fragments from global: rows kg, kg+1 of W (lane-coalesced in N)
            const int kg = k0 + kk + koff;
            const float* wr0 = W + (size_t)kg * N;
            const float* wr1 = wr0 + N;
            v2f b0; b0.x = wr0[nc];      b0.y = wr1[nc];
            v2f b1; b1.x = wr0[nc + 16]; b1.y = wr1[nc + 16];
            v2f b2; b2.x = wr0[nc + 32]; b2.y = wr1[nc + 32];
            v2f b3; b3.x = wr0[nc + 48]; b3.y = wr1[nc + 48];
            acc0 = wmma_f32(a, b0, acc0);
            acc1 = wmma_f32(a, b1, acc1);
            acc2 = wmma_f32(a, b2, acc2);
            acc3 = wmma_f32(a, b3, acc3);
        }
        __syncthreads();
    }

    // C/D layout: VGPR g, lanes 0-15 -> M=g, lanes 16-31 -> M=g+8; N = lane&15
    const int cm = m0 + ((lane < 16) ? 0 : 8);
    const int cn = nw + (lane & 15);
    #pragma unroll
    for (int g = 0; g < 8; ++g) {
        float* crow = C + (size_t)(cm + g) * N + cn;
        crow[0]  = acc0[g];
        crow[16] = acc1[g];
        crow[32] = acc2[g];
        crow[48] = acc3[g];
    }
}

// ---------------------------------------------------------------------------
// RoPE: t is (128 x ncols) with column n = head*128 + d; pos = START + (row%16)
// ---------------------------------------------------------------------------
__global__ void rope_kernel(float* __restrict__ t,
                            const float* __restrict__ cosb,
                            const float* __restrict__ sinb,
                            int ncols) {
    const int i = blockIdx.x * blockDim.x + threadIdx.x;     // pair index
    const int pairs_per_row = ncols >> 1;
    const int total = MROWS * pairs_per_row;
    if (i >= total) return;
    const int row = i / pairs_per_row;
    const int p   = i - row * pairs_per_row;
    const int d2  = p & 63;                                  // (d % 128) / 2
    const int s   = row & (SEQ - 1);
    const float c = cosb[(size_t)(STARTP + s) * (HD / 2) + d2];
    const float sn = sinb[(size_t)(STARTP + s) * (HD / 2) + d2];
    float* tp = t + (size_t)row * ncols + 2 * p;
    const float t0 = tp[0], t1 = tp[1];
    tp[0] = t0 * c - t1 * sn;
    tp[1] = t0 * sn + t1 * c;
}

// ---------------------------------------------------------------------------
// Scores: S[bh][s][j] = (q_h . k_j) / sqrt(128), one wave per (bh, 16-key tile)
// K-tile staged coalesced in LDS, B fragment read transposed (stride 132:
// bank-conflict-free, 16B-aligned rows). grid = (KLEN/16, B*H), block = 32.
// ---------------------------------------------------------------------------
__global__ void scores_kernel(const float* __restrict__ q,
                              const float* __restrict__ cache_k,
                              const float* __restrict__ k_new,
                              float* __restrict__ scores) {
    __shared__ alignas(16) float Ks[16][HD + 4];   // 16 x 132 floats

    const int lane = threadIdx.x;
    const int bh = blockIdx.y;
    const int b = bh >> 5;               // / H
    const int h = bh & (Hh - 1);
    const int kvh = h >> 2;              // H/KVH = 4
    const int j0 = blockIdx.x * 16;

    // stage 16 key rows (coalesced float4): lane loads one float4 per row
    for (int r = 0; r < 16; ++r) {
        const int j = j0 + r;
        const float* src = (j < STARTP)
            ? cache_k + ((size_t)(b * MAXLEN + j) * KVH + kvh) * HD
            : k_new   + ((size_t)(b * SEQ + (j - STARTP)) * KVH + kvh) * HD;
        const float4 v = *reinterpret_cast<const float4*>(src + lane * 4);
        *reinterpret_cast<float4*>(&Ks[r][lane * 4]) = v;
    }
    __syncthreads();

    const int mrow = lane & 15;
    const int koff = (lane >> 4) * 2;
    const float* qh = q + (size_t)(b * SEQ) * Dm + h * HD;   // q rows for this (b,h)

    v8f acc = {};
    #pragma unroll
    for (int kk = 0; kk < HD; kk += 4) {
        const float2 af = *reinterpret_cast<const float2*>(
            qh + (size_t)mrow * Dm + kk + koff);
        v2f a; a.x = af.x; a.y = af.y;
        // B[k=d][n=j] = K[j0 + (lane&15)][d]  (transposed read from LDS)
        v2f bf;
        bf.x = Ks[mrow][kk + koff];
        bf.y = Ks[mrow][kk + koff + 1];
        acc = wmma_f32(a, bf, acc);
    }

    const float scale = 0.08838834764831845f;   // 1/sqrt(128)
    const size_t sbase = (size_t)bh * SEQ * KLEN;
    const int sm = (lane < 16) ? 0 : 8;
    const int sj = j0 + (lane & 15);
    #pragma unroll
    for (int g = 0; g < 8; ++g)
        scores[sbase + (size_t)(sm + g) * KLEN + sj] = acc[g] * scale;
}

// ---------------------------------------------------------------------------
// Softmax over KLEN, one wave per row (B*H*SEQ = 4096 rows). wave32 shuffles.
// ---------------------------------------------------------------------------
__global__ void softmax_kernel(float* __restrict__ scores) {
    const int row = blockIdx.x;
    const int lane = threadIdx.x;
    float* p = scores + (size_t)row * KLEN;

    float mx = -3.402823466e38f;
    for (int j = lane; j < KLEN; j += 32) mx = fmaxf(mx, p[j]);
    #pragma unroll
    for (int o = 16; o > 0; o >>= 1) mx = fmaxf(mx, __shfl_xor(mx, o, 32));

    float sum = 0.0f;
    for (int j = lane; j < KLEN; j += 32) {
        const float e = __expf(p[j] - mx);
        p[j] = e;
        sum += e;
    }
    #pragma unroll
    for (int o = 16; o > 0; o >>= 1) sum += __shfl_xor(sum, o, 32);

    const float inv = 1.0f / sum;
    for (int j = lane; j < KLEN; j += 32) p[j] *= inv;
}

// ---------------------------------------------------------------------------
// out[s][d] = sum_j P[s][j] * V[j][d]  per (b,h). One wave per (bh, 16-d tile).
// A = probs (contiguous K pairs), B = V rows (lane-coalesced along d).
// grid = (HD/16, B*H), block = 32. Writes attn as (128 x 4096) matrix.
// ---------------------------------------------------------------------------
__global__ void pv_kernel(const float* __restrict__ scores,
                          const float* __restrict__ cache_v,
                          const float* __restrict__ v_new,
                          float* __restrict__ attn) {
    const int lane = threadIdx.x;
    const int bh = blockIdx.y;
    const int b = bh >> 5;
    const int h = bh & (Hh - 1);
    const int kvh = h >> 2;
    const int n0 = blockIdx.x * 16;

    const int mrow = lane & 15;
    const int koff = (lane >> 4) * 2;
    const int n = n0 + mrow;
    const float* prow = scores + ((size_t)bh * SEQ + mrow) * KLEN;

    v8f acc = {};
    for (int j = 0; j < KLEN; j += 4) {
        const int ja = j + koff;                 // ja and ja+1 are on same side of START
        const float2 af = *reinterpret_cast<const float2*>(prow + ja);
        v2f a; a.x = af.x; a.y = af.y;
        const float* vp = (ja < STARTP)
            ? cache_v + ((size_t)(b * MAXLEN + ja) * KVH + kvh) * HD
            : v_new   + ((size_t)(b * SEQ + (ja - STARTP)) * KVH + kvh) * HD;
        v2f bf;
        bf.x = vp[n];
        bf.y = vp[(size_t)KVH * HD + n];         // next j row (stride 1024 floats)
        acc = wmma_f32(a, bf, acc);
    }

    // attn[b][s][h][d] laid out as (128 x 4096): row = b*16+s, col = h*128+d
    const int sm = (lane < 16) ? 0 : 8;
    const int cn = h * HD + n0 + (lane & 15);
    #pragma unroll
    for (int g = 0; g < 8; ++g)
        attn[(size_t)(b * SEQ + sm + g) * Dm + cn] = acc[g];
}

// ---------------------------------------------------------------------------
// Launcher
// inputs: x, wq, wk, wv, wo, cos, sin, cache_k, cache_v, start_pos(=1024)
// ---------------------------------------------------------------------------
extern "C" void kernel_launch(void* const* d_in, const int* in_sizes, int n_in,
                              void* d_out, int out_size, void* d_ws, size_t ws_size,
                              hipStream_t stream) {
    const float* x       = (const float*)d_in[0];
    const float* wq      = (const float*)d_in[1];
    const float* wk      = (const float*)d_in[2];
    const float* wv      = (const float*)d_in[3];
    const float* wo      = (const float*)d_in[4];
    const float* cosb    = (const float*)d_in[5];
    const float* sinb    = (const float*)d_in[6];
    const float* cache_k = (const float*)d_in[7];
    const float* cache_v = (const float*)d_in[8];
    float* out = (float*)d_out;

    // workspace layout (floats): total ~5.57M floats (~21.3 MB)
    float* ws  = (float*)d_ws;
    float* qws = ws;                                   // 128*4096
    float* kws = qws + (size_t)MROWS * Dm;             // 128*1024
    float* vws = kws + (size_t)MROWS * (KVH * HD);     // 128*1024
    float* sws = vws + (size_t)MROWS * (KVH * HD);     // 256*16*1040
    float* aws = sws + (size_t)Bb * Hh * SEQ * KLEN;   // 128*4096

    const dim3 gblk(128);

    // QKV projections
    gemm128_wmma<<<dim3(Dm / 256, MROWS / 16), gblk, 0, stream>>>(x, wq, qws, Dm, Dm);
    gemm128_wmma<<<dim3((KVH * HD) / 256, MROWS / 16), gblk, 0, stream>>>(x, wk, kws, Dm, KVH * HD);
    gemm128_wmma<<<dim3((KVH * HD) / 256, MROWS / 16), gblk, 0, stream>>>(x, wv, vws, Dm, KVH * HD);

    // RoPE on q and new k
    {
        const int qp = MROWS * Dm / 2;
        const int kp = MROWS * (KVH * HD) / 2;
        rope_kernel<<<(qp + 255) / 256, 256, 0, stream>>>(qws, cosb, sinb, Dm);
        rope_kernel<<<(kp + 255) / 256, 256, 0, stream>>>(kws, cosb, sinb, KVH * HD);
    }

    // attention
    scores_kernel<<<dim3(KLEN / 16, Bb * Hh), 32, 0, stream>>>(qws, cache_k, kws, sws);
    softmax_kernel<<<Bb * Hh * SEQ, 32, 0, stream>>>(sws);
    pv_kernel<<<dim3(HD / 16, Bb * Hh), 32, 0, stream>>>(sws, cache_v, vws, aws);

    // output projection
    gemm128_wmma<<<dim3(Dm / 256, MROWS / 16), gblk, 0, stream>>>(aws, wo, out, Dm, Dm);
}